// BidirectionalAttention_79474074845241
// MI455X (gfx1250) — compile-verified
//
#include <hip/hip_runtime.h>

#define Bn 64
#define Cn 2048
#define Qn 64
#define En 256
#define CT 128      // context rows per workgroup (8 waves x 16 rows)
#define SPLIT 8     // C-splits for q2c partial reduction

typedef __attribute__((ext_vector_type(2))) float v2f;
typedef __attribute__((ext_vector_type(8))) float v8f;

// ---------------------------------------------------------------------------
// Phase 1: per (batch, 128-row context tile):
//   sim = (ctx .* w_m) @ q^T + c.w_c + q.w_q      (f32 WMMA 16x16x4)
//   mrow[b,c] = max_q sim                         (for query2context)
//   c2q = softmax_q(sim) @ q                      (f32 WMMA 16x16x4)
// ---------------------------------------------------------------------------
__launch_bounds__(256)
__global__ void bidaf_sim_c2q(const float* __restrict__ ctx,
                              const float* __restrict__ qst,
                              const float* __restrict__ wq,
                              const float* __restrict__ wc,
                              const float* __restrict__ wm,
                              float* __restrict__ c2q,
                              float* __restrict__ mrow) {
  __shared__ float q_lds[Qn * En];   // 64 KB: question tile, row-major [q][e]
  __shared__ float p_lds[CT * Qn];   // 32 KB: softmax probabilities
  __shared__ float wm_lds[En];
  __shared__ float wc_lds[En];
  __shared__ float wq_lds[En];
  __shared__ float qw_lds[Qn];
  __shared__ float cw_lds[CT];

  const int tid  = threadIdx.x;
  const int wave = tid >> 5;
  const int lane = tid & 31;
  const int lh   = lane & 15;   // lane within half
  const int hi   = lane >> 4;   // which half of the wave

  const int blk = blockIdx.x;
  const int b   = blk / (Cn / CT);
  const int c0  = (blk % (Cn / CT)) * CT;

  // ---- stage question tile + weight vectors into LDS ----
  const float* qbase = qst + (size_t)b * Qn * En;
  for (int i = tid; i < (Qn * En) / 4; i += 256)
    ((float4*)q_lds)[i] = ((const float4*)qbase)[i];
  for (int i = tid; i < En; i += 256) {
    wm_lds[i] = wm[i];
    wc_lds[i] = wc[i];
    wq_lds[i] = wq[i];
  }
  __syncthreads();

  // q_w[q] = question[q] . w_question
  if (tid < Qn) {
    float s = 0.f;
    for (int e = 0; e < En; ++e) s += q_lds[tid * En + e] * wq_lds[e];
    qw_lds[tid] = s;
  }
  __syncthreads();

  // ---- GEMM1: S(16xQn per wave) = (ctx .* wm) @ q^T ----
  v8f acc[4];
  {
    v8f z = {0.f,0.f,0.f,0.f,0.f,0.f,0.f,0.f};
#pragma unroll
    for (int t = 0; t < 4; ++t) acc[t] = z;
  }
  const int rowg = c0 + wave * 16 + lh;                 // A-frag row (M = lane&15)
  const float* abase = ctx + ((size_t)b * Cn + rowg) * En;
  float cwp = 0.f;                                      // partial ctx . w_c
  for (int k = 0; k < En; k += 4) {
    const int kk = k + 2 * hi;                          // A layout: hi half holds K+2,K+3
    v2f araw = *(const v2f*)(abase + kk);
    cwp += araw.x * wc_lds[kk] + araw.y * wc_lds[kk + 1];
    v2f a;
    a.x = araw.x * wm_lds[kk];
    a.y = araw.y * wm_lds[kk + 1];
    v2f b0 = *(const v2f*)(q_lds + (size_t)(0  + lh) * En + kk);
    v2f b1 = *(const v2f*)(q_lds + (size_t)(16 + lh) * En + kk);
    v2f b2 = *(const v2f*)(q_lds + (size_t)(32 + lh) * En + kk);
    v2f b3 = *(const v2f*)(q_lds + (size_t)(48 + lh) * En + kk);
    acc[0] = __builtin_amdgcn_wmma_f32_16x16x4_f32(false, a, false, b0, (short)0, acc[0], false, false);
    acc[1] = __builtin_amdgcn_wmma_f32_16x16x4_f32(false, a, false, b1, (short)0, acc[1], false, false);
    acc[2] = __builtin_amdgcn_wmma_f32_16x16x4_f32(false, a, false, b2, (short)0, acc[2], false, false);
    acc[3] = __builtin_amdgcn_wmma_f32_16x16x4_f32(false, a, false, b3, (short)0, acc[3], false, false);
  }
  // reduce c_w across the two half-lanes holding the same row
  cwp += __shfl_xor(cwp, 16, 32);
  if (hi == 0) cw_lds[wave * 16 + lh] = cwp;
  __syncthreads();

  // ---- bias + row softmax directly on the WMMA D layout ----
  // acc[t][r] holds sim[row = wave*16 + 8*hi + r, col = 16*t + lh]
  float rmax[8];
#pragma unroll
  for (int r = 0; r < 8; ++r) {
    const float cwr = cw_lds[wave * 16 + 8 * hi + r];
    float mx = -3.0e38f;
#pragma unroll
    for (int t = 0; t < 4; ++t) {
      float v = acc[t][r] + cwr + qw_lds[16 * t + lh];
      acc[t][r] = v;
      mx = fmaxf(mx, v);
    }
    mx = fmaxf(mx, __shfl_xor(mx, 1, 32));
    mx = fmaxf(mx, __shfl_xor(mx, 2, 32));
    mx = fmaxf(mx, __shfl_xor(mx, 4, 32));
    mx = fmaxf(mx, __shfl_xor(mx, 8, 32));
    rmax[r] = mx;
  }
  if (lh == 0) {
#pragma unroll
    for (int r = 0; r < 8; ++r)
      mrow[(size_t)b * Cn + c0 + wave * 16 + 8 * hi + r] = rmax[r];
  }
#pragma unroll
  for (int r = 0; r < 8; ++r) {
    float s = 0.f;
#pragma unroll
    for (int t = 0; t < 4; ++t) {
      float e = __expf(acc[t][r] - rmax[r]);
      acc[t][r] = e;
      s += e;
    }
    s += __shfl_xor(s, 1, 32);
    s += __shfl_xor(s, 2, 32);
    s += __shfl_xor(s, 4, 32);
    s += __shfl_xor(s, 8, 32);
    const float inv = 1.0f / s;
#pragma unroll
    for (int t = 0; t < 4; ++t)
      p_lds[(size_t)(wave * 16 + 8 * hi + r) * Qn + 16 * t + lh] = acc[t][r] * inv;
  }
  __syncthreads();

  // ---- GEMM2: c2q(16xEn per wave) = P @ question, E in chunks of 64 ----
  float* obase = c2q + ((size_t)b * Cn + c0 + wave * 16) * En;
#pragma unroll 1
  for (int ec = 0; ec < 4; ++ec) {
    v8f d[4];
    {
      v8f z = {0.f,0.f,0.f,0.f,0.f,0.f,0.f,0.f};
#pragma unroll
      for (int t = 0; t < 4; ++t) d[t] = z;
    }
    for (int k = 0; k < Qn; k += 4) {
      const int kk = k + 2 * hi;
      v2f a2 = *(const v2f*)(p_lds + (size_t)(wave * 16 + lh) * Qn + kk);
#pragma unroll
      for (int t = 0; t < 4; ++t) {
        const int n = ec * 64 + 16 * t + lh;
        v2f b2;
        b2.x = q_lds[(size_t)kk * En + n];
        b2.y = q_lds[(size_t)(kk + 1) * En + n];
        d[t] = __builtin_amdgcn_wmma_f32_16x16x4_f32(false, a2, false, b2, (short)0, d[t], false, false);
      }
    }
#pragma unroll
    for (int t = 0; t < 4; ++t)
#pragma unroll
      for (int r = 0; r < 8; ++r)
        obase[(size_t)(8 * hi + r) * En + ec * 64 + 16 * t + lh] = d[t][r];
  }
}

// ---------------------------------------------------------------------------
// Phase 2: beta[b,:] = softmax_c(mrow[b,:])
// ---------------------------------------------------------------------------
__launch_bounds__(256)
__global__ void bidaf_beta(const float* __restrict__ mrow, float* __restrict__ beta) {
  const int b = blockIdx.x;
  const int tid = threadIdx.x;
  __shared__ float red[8];
  const float* mb = mrow + (size_t)b * Cn;

  float mx = -3.0e38f;
  for (int c = tid; c < Cn; c += 256) mx = fmaxf(mx, mb[c]);
  for (int o = 1; o < 32; o <<= 1) mx = fmaxf(mx, __shfl_xor(mx, o, 32));
  if ((tid & 31) == 0) red[tid >> 5] = mx;
  __syncthreads();
  float bm = red[0];
#pragma unroll
  for (int i = 1; i < 8; ++i) bm = fmaxf(bm, red[i]);
  __syncthreads();

  float sum = 0.f;
  for (int c = tid; c < Cn; c += 256) sum += __expf(mb[c] - bm);
  for (int o = 1; o < 32; o <<= 1) sum += __shfl_xor(sum, o, 32);
  if ((tid & 31) == 0) red[tid >> 5] = sum;
  __syncthreads();
  float ts = 0.f;
#pragma unroll
  for (int i = 0; i < 8; ++i) ts += red[i];
  const float inv = 1.0f / ts;

  float* bb = beta + (size_t)b * Cn;
  for (int c = tid; c < Cn; c += 256) bb[c] = __expf(mb[c] - bm) * inv;
}

// ---------------------------------------------------------------------------
// Phase 3a: partial q2c[b,e] over a C-slice (coalesced context reread)
// ---------------------------------------------------------------------------
__launch_bounds__(256)
__global__ void bidaf_q2c_part(const float* __restrict__ ctx,
                               const float* __restrict__ beta,
                               float* __restrict__ part) {
  const int blk = blockIdx.x;
  const int b = blk / SPLIT;
  const int s = blk % SPLIT;
  const int e = threadIdx.x;                 // 256 threads == En
  const int cbeg = s * (Cn / SPLIT);
  const float* base = ctx + ((size_t)b * Cn + cbeg) * En + e;
  const float* bb = beta + (size_t)b * Cn + cbeg;
  float acc = 0.f;
  for (int c = 0; c < Cn / SPLIT; ++c)
    acc += bb[c] * base[(size_t)c * En];
  part[((size_t)b * SPLIT + s) * En + e] = acc;
}

// ---------------------------------------------------------------------------
// Phase 3b: combine partials into q2c slice of d_out
// ---------------------------------------------------------------------------
__launch_bounds__(256)
__global__ void bidaf_q2c_final(const float* __restrict__ part, float* __restrict__ q2c) {
  const int b = blockIdx.x;
  const int e = threadIdx.x;
  float s = 0.f;
#pragma unroll
  for (int i = 0; i < SPLIT; ++i)
    s += part[((size_t)b * SPLIT + i) * En + e];
  q2c[(size_t)b * En + e] = s;
}

extern "C" void kernel_launch(void* const* d_in, const int* in_sizes, int n_in,
                              void* d_out, int out_size, void* d_ws, size_t ws_size,
                              hipStream_t stream) {
  (void)in_sizes; (void)n_in; (void)out_size; (void)ws_size;
  const float* ctx = (const float*)d_in[0];   // (B, C, E)
  const float* qst = (const float*)d_in[1];   // (B, Q, E)
  const float* wq  = (const float*)d_in[2];   // (E,)
  const float* wc  = (const float*)d_in[3];   // (E,)
  const float* wm  = (const float*)d_in[4];   // (E,)

  float* out  = (float*)d_out;                         // c2q (B*C*E) ++ q2c (B*E)
  float* q2c  = out + (size_t)Bn * Cn * En;

  float* wsf  = (float*)d_ws;
  float* mrow = wsf;                                   // B*C
  float* beta = wsf + (size_t)Bn * Cn;                 // B*C
  float* part = wsf + (size_t)2 * Bn * Cn;             // B*SPLIT*E

  bidaf_sim_c2q<<<Bn * (Cn / CT), 256, 0, stream>>>(ctx, qst, wq, wc, wm, out, mrow);
  bidaf_beta<<<Bn, 256, 0, stream>>>(mrow, beta);
  bidaf_q2c_part<<<Bn * SPLIT, 256, 0, stream>>>(ctx, beta, part);
  bidaf_q2c_final<<<Bn, 256, 0, stream>>>(part, q2c);
}